// proteinEncoder_51453708206755
// MI455X (gfx1250) — compile-verified
//
#include <hip/hip_runtime.h>
#include <hip/hip_bf16.h>
#include <stdint.h>
#include <stddef.h>

#define N_NODES 50000
#define N_EDGES 800000

typedef __attribute__((ext_vector_type(2))) float v2f;
typedef __attribute__((ext_vector_type(8))) float v8f;

// ---------------- degree / normalization ----------------
__global__ void k_init_deg(float* __restrict__ deg, int n) {
    int i = blockIdx.x * blockDim.x + threadIdx.x;
    if (i < n) deg[i] = 1.0f;   // self-loop contributes 1 to every node's degree
}

__global__ void k_count_deg(const long long* __restrict__ dst,
                            float* __restrict__ deg, int ne) {
    int e = blockIdx.x * blockDim.x + threadIdx.x;
    if (e < ne) atomicAdd(&deg[(int)dst[e]], 1.0f);
}

__global__ void k_dinv(float* __restrict__ deg, int n) {
    int i = blockIdx.x * blockDim.x + threadIdx.x;
    if (i < n) deg[i] = rsqrtf(deg[i]);   // deg >= 1 always (self-loop)
}

// ---------------- W repack into WMMA-fragment-major layout ----------------
// Bp[((k/8)*2 + h)*N*4 + n*4 + j] = B[k', n] where
//   j=0: k'=8c+2h   j=1: k'=8c+2h+1   j=2: k'=8c+4+2h   j=3: k'=8c+4+2h+1
// so a lane (half h, column n) reads the B values for two consecutive
// 16x16x4 WMMAs (k-bases 8c and 8c+4) with a single b128 load.
template<int K, int N>
__global__ void k_pack_B(const float* __restrict__ B, float* __restrict__ Bp) {
    int i = blockIdx.x * blockDim.x + threadIdx.x;
    if (i >= K * N) return;
    int j     = i & 3;
    int n     = (i >> 2) % N;
    int rest  = (i >> 2) / N;       // chunk*2 + h
    int h     = rest & 1;
    int chunk = rest >> 1;
    int k     = chunk * 8 + (j >> 1) * 4 + h * 2 + (j & 1);
    Bp[i] = B[k * N + n];
}

// ---------------- fp32 WMMA GEMM: C[M x N] = A[:, acol:acol+K] * Bp ---------
// Block = 256 threads = 8 wave32s, one 16-row M tile per block.
// A tile staged in LDS via float4 (row pad +4 floats -> bank-conflict-free).
template<int K, int N, int TPW>
__global__ __launch_bounds__(256) void k_gemm_wmma(
        const float* __restrict__ A, int lda, int acol,
        const float* __restrict__ Bp, float* __restrict__ C) {
    constexpr int LDSK = K + 4;
    __shared__ float As[16 * LDSK];

    const int mbase = blockIdx.x * 16;
    const int tid   = threadIdx.x;

    // cooperative load of the 16 x K A tile, 128-bit granularity
    for (int idx = tid; idx < 16 * (K / 4); idx += 256) {
        int r = idx / (K / 4), c4 = (idx % (K / 4)) * 4;
        float4 v = *(const float4*)(A + (size_t)(mbase + r) * lda + acol + c4);
        *(float4*)(&As[r * LDSK + c4]) = v;
    }
    __syncthreads();

    const int wave = tid >> 5;      // wave32
    const int lane = tid & 31;
    const int m    = lane & 15;
    const int h    = lane >> 4;          // 0 or 1
    const int kh   = h << 1;             // lanes 0-15: K=0,1  lanes 16-31: K=2,3

    for (int t = 0; t < TPW; ++t) {
        const int nbase = (wave * TPW + t) * 16;
        const int n     = nbase + (lane & 15);
        v8f acc = {0.f, 0.f, 0.f, 0.f, 0.f, 0.f, 0.f, 0.f};
        #pragma unroll 4
        for (int k = 0; k < K; k += 8) {
            v2f a0, a1;
            a0.x = As[m * LDSK + k + kh];
            a0.y = As[m * LDSK + k + kh + 1];
            a1.x = As[m * LDSK + k + 4 + kh];
            a1.y = As[m * LDSK + k + 4 + kh + 1];
            float4 bq = *(const float4*)(Bp + (((size_t)(k >> 3) * 2 + h) * N + n) * 4);
            v2f b0; b0.x = bq.x; b0.y = bq.y;
            v2f b1; b1.x = bq.z; b1.y = bq.w;
            acc = __builtin_amdgcn_wmma_f32_16x16x4_f32(
                      false, a0, false, b0, (short)0, acc, false, false);
            acc = __builtin_amdgcn_wmma_f32_16x16x4_f32(
                      false, a1, false, b1, (short)0, acc, false, false);
        }
        // C/D layout: VGPR r -> M = r (lanes 0-15) / r+8 (lanes 16-31)
        const int mrow = mbase + (h << 3);
        #pragma unroll
        for (int r = 0; r < 8; ++r)
            C[(size_t)(mrow + r) * N + n] = acc[r];
    }
}

// ------- seed accumulator with self-loop + bias: out = xw*dinv^2 + bias ------
template<int F>
__global__ void k_seed(const float* __restrict__ xw,
                       const float* __restrict__ dinv,
                       const float* __restrict__ bias,
                       float* __restrict__ out, int n_nodes) {
    int i = blockIdx.x * blockDim.x + threadIdx.x;
    if (i >= n_nodes * F) return;
    int node = i / F, f = i % F;
    float dd = dinv[node];
    out[i] = xw[i] * (dd * dd) + bias[f];
}

// ---------------- edge scatter:  out[dst] += xw[src] * dinv[src]*dinv[dst] ---
// F/4 threads per edge; each thread gathers a float4 and issues 4 f32 atomics.
template<int F>
__global__ void k_scatter(const float* __restrict__ xw,
                          const float* __restrict__ dinv,
                          const long long* __restrict__ src,
                          const long long* __restrict__ dst,
                          float* __restrict__ out, int ne) {
    constexpr int TPE = F / 4;                 // 64 (F=256) or 32 (F=128)
    int tid = blockIdx.x * blockDim.x + threadIdx.x;
    int e   = tid / TPE;
    int f4  = (tid % TPE) * 4;
    if (e >= ne) return;
    int s = (int)src[e], d = (int)dst[e];
    float nrm = dinv[s] * dinv[d];
    const float4 v = *(const float4*)(xw + (size_t)s * F + f4);
    float* o = out + (size_t)d * F + f4;
    atomicAdd(o + 0, v.x * nrm);
    atomicAdd(o + 1, v.y * nrm);
    atomicAdd(o + 2, v.z * nrm);
    atomicAdd(o + 3, v.w * nrm);
}

__global__ void k_relu(float* __restrict__ p, int n) {
    int i = blockIdx.x * blockDim.x + threadIdx.x;
    if (i < n) p[i] = fmaxf(p[i], 0.0f);
}

// ---------------- driver ----------------
extern "C" void kernel_launch(void* const* d_in, const int* in_sizes, int n_in,
                              void* d_out, int out_size, void* d_ws, size_t ws_size,
                              hipStream_t stream) {
    const float*     x   = (const float*)d_in[0];       // (50000, 1100) fp32
    const long long* ei  = (const long long*)d_in[1];   // (2, 800000) int64
    const float*     W1  = (const float*)d_in[2];       // (512, 256)
    const float*     b1  = (const float*)d_in[3];       // (256,)
    const float*     W2  = (const float*)d_in[4];       // (256, 128)
    const float*     b2  = (const float*)d_in[5];       // (128,)
    float*           out = (float*)d_out;               // (50000, 128)

    // workspace layout (256B-aligned offsets)
    char*  ws   = (char*)d_ws;
    float* dinv = (float*)ws;                                   // 50000 f32
    float* xw   = (float*)(ws + 200192);                        // 50000*256 f32
    float* h    = (float*)(ws + 200192 + (size_t)51200000);     // 50000*256 f32
    float* Bp   = (float*)(ws + 200192 + (size_t)102400000);    // 512*256 f32 (reused)
    // layer-2 xw (50000*128) aliases xw — it is dead after k_seed<256>+scatter.
    const long long* src = ei;
    const long long* dst = ei + N_EDGES;

    // symmetric normalization (shared by both layers)
    k_init_deg <<<(N_NODES + 255) / 256, 256, 0, stream>>>(dinv, N_NODES);
    k_count_deg<<<(N_EDGES + 255) / 256, 256, 0, stream>>>(dst, dinv, N_EDGES);
    k_dinv     <<<(N_NODES + 255) / 256, 256, 0, stream>>>(dinv, N_NODES);

    // ---- layer 1: xw = x[:,588:1100] @ W1 ; h = relu(Aggr(xw) + b1) ----
    k_pack_B<512, 256><<<(512 * 256 + 255) / 256, 256, 0, stream>>>(W1, Bp);
    k_gemm_wmma<512, 256, 2><<<N_NODES / 16, 256, 0, stream>>>(x, 1100, 588, Bp, xw);
    k_seed<256><<<(N_NODES * 256 + 255) / 256, 256, 0, stream>>>(xw, dinv, b1, h, N_NODES);
    k_scatter<256><<<(N_EDGES * 64 + 255) / 256, 256, 0, stream>>>(xw, dinv, src, dst, h, N_EDGES);
    k_relu<<<(N_NODES * 256 + 255) / 256, 256, 0, stream>>>(h, N_NODES * 256);

    // ---- layer 2: xw2 = h @ W2 ; out = Aggr(xw2) + b2 ----
    float* xw2 = xw;
    k_pack_B<256, 128><<<(256 * 128 + 255) / 256, 256, 0, stream>>>(W2, Bp);
    k_gemm_wmma<256, 128, 1><<<N_NODES / 16, 256, 0, stream>>>(h, 256, 0, Bp, xw2);
    k_seed<128><<<(N_NODES * 128 + 255) / 256, 256, 0, stream>>>(xw2, dinv, b2, out, N_NODES);
    k_scatter<128><<<(N_EDGES * 32 + 255) / 256, 256, 0, stream>>>(xw2, dinv, src, dst, out, N_EDGES);
}